// CARAFE_17910013624967
// MI455X (gfx1250) — compile-verified
//
#include <hip/hip_runtime.h>

// CARAFE upsampling, fixed shapes: B=4, C=256, H=W=64, k=5, ratio=2.
// out[b,c,2h+py,2w+px] = sum_{ki,kj} x_pad[b,c,h+ki,w+kj] * kernel[b,ki*5+kj,2h+py,2w+px]
//
// WMMA mapping (V_WMMA_F32_16X16X4_F32, wave32):
//   M = 16 channels, N = 16 output pixels (4 coarse pixels x 4 subpos),
//   K = 100 = 25 taps x 4 coarse pixels, tap-major (K = 4t + cpl).
//   A[m,K] = x patch (dense, LDS, compile-const offsets).
//   B[K,n] = block-diagonal weights, pre-expanded (pre-zeroed) into LDS so the
//            inner loop is: 1 ds_load pair (A) + 1 ds_load_b64 (B) + 1 wmma.
// CDNA5 paths: x tile staged with GLOBAL_LOAD_ASYNC_TO_LDS_B64 (ASYNCcnt) +
//   s_wait_asynccnt; dynamic LDS (80,640 B) leveraging the 320 KB WGP pool.

typedef __attribute__((ext_vector_type(2))) float v2f;
typedef __attribute__((ext_vector_type(8))) float v8f;

#define B_    4
#define C_    256
#define H_    64
#define W_    64
#define KS    5
#define R_    2
#define OH_   (H_ * R_)        // 128
#define OW_   (W_ * R_)        // 128
#define K2    (KS * KS)        // 25
#define TBCP  8                // coarse pixels per block (along w)
#define XW    (TBCP + KS - 1)  // 12 staged x columns, window [w0-2, w0+9]
#define CHSTR (KS * XW)        // 60 dwords/channel; stride 60 -> conflict-free A loads
#define BCOL  48               // padded B columns: hi-stride 96 dwords -> bank sets
                               // {2n} vs {32+2n}, disjoint across lane halves
#define XS_FLOATS  (C_ * CHSTR)          // 15360
#define WSB_FLOATS (K2 * 2 * BCOL * 2)   // 4800
#define SMEM_BYTES ((XS_FLOATS + WSB_FLOATS) * 4)  // 80,640 B (dynamic LDS)
#define NTHR  512              // 16 waves; one 16-channel slice per wave

__global__ __launch_bounds__(NTHR)
void carafe_wmma_async(const float* __restrict__ x,
                       const float* __restrict__ kern,
                       float* __restrict__ out) {
  extern __shared__ __align__(16) float smem[];
  float* xs  = smem;                 // [256][5][12]
  float* wsB = smem + XS_FLOATS;     // [25][2(hi)][48 cols][2(j)]

  const int tid = threadIdx.x;
  const int w0  = blockIdx.x * TBCP;
  const int h0  = blockIdx.y;
  const int b   = blockIdx.z;

  const float* xb = x + (size_t)b * C_ * H_ * W_;

  // ---- async-stage x tile: [256 ch] x [5 rows] x [12 w], B64 per lane ----
  // Clamped addresses (clamp bounds even -> 8B alignment preserved); OOB halo
  // entries are zero-fixed after the wait.
  for (int i = tid; i < C_ * KS * (XW / 2); i += NTHR) {
    int c   = i / (KS * (XW / 2));
    int rem = i - c * (KS * (XW / 2));
    int row = rem / (XW / 2);
    int q   = rem - row * (XW / 2);
    int wl  = 2 * q;                                  // 0,2,...,10
    int hh  = h0 - 2 + row;
    hh = hh < 0 ? 0 : (hh > H_ - 1 ? H_ - 1 : hh);
    int ww  = w0 - 2 + wl;                            // even
    ww = ww < 0 ? 0 : (ww > W_ - 2 ? W_ - 2 : ww);    // stays even
    unsigned ldsoff = (unsigned)(size_t)&xs[c * CHSTR + row * XW + wl];
    const float* gp = xb + ((size_t)c * H_ + hh) * W_ + ww;
    asm volatile("global_load_async_to_lds_b64 %0, %1, off"
                 :: "v"(ldsoff), "v"(gp) : "memory");
  }

  // ---- stage block-diagonal B, pre-expanded & pre-zeroed ----
  // wsB[((t*2+hi)*BCOL + col)*2 + j] = (2*hi+j == cpl(col)) ? weight(t, col) : 0
  // col = g*16 + n, n = py*8 + cpl*2 + px.
  const float* kb = kern + (size_t)b * K2 * OH_ * OW_;
  for (int i = tid; i < K2 * 2 * 32 * 2; i += NTHR) {   // 3200 valid entries
    int t    = i >> 7;          // 128 entries per tap
    int rem  = i & 127;
    int hi   = rem >> 6;
    int rem2 = rem & 63;
    int col  = rem2 >> 1;       // 0..31 (= g*16 + n)
    int j    = rem2 & 1;
    int row  = 2 * hi + j;      // K-row within chunk
    int n    = col & 15;
    int g    = col >> 4;
    int py   = (n >> 3) & 1;
    int cpl  = (n >> 1) & 3;
    int px   = n & 1;
    float v = 0.0f;
    if (row == cpl)
      v = kb[(t * OH_ + R_ * h0 + py) * OW_ + R_ * (w0 + 4 * g + cpl) + px];
    wsB[((t * 2 + hi) * BCOL + col) * 2 + j] = v;
  }

  asm volatile("s_wait_asynccnt 0x0" ::: "memory");
  __syncthreads();

  // ---- zero the out-of-bounds x halo (block-uniform branch; edge blocks only) ----
  const bool interior = (h0 >= 2) && (h0 <= H_ - 3) &&
                        (w0 >= 2) && (w0 + XW - 2 <= W_ - 1);
  if (!interior) {
    for (int i = tid; i < C_ * CHSTR; i += NTHR) {
      int c   = i / CHSTR;
      int rem = i - c * CHSTR;
      int row = rem / XW;
      int wl  = rem - row * XW;
      int hh  = h0 - 2 + row;
      int ww  = w0 - 2 + wl;
      if (hh < 0 || hh >= H_ || ww < 0 || ww >= W_) xs[i] = 0.0f;
    }
  }
  __syncthreads();

  // ---- WMMA compute: per chunk = 2 A dwords + 1 B b64 + 1 wmma ----
  const int wvid = tid >> 5;         // 16-channel slice per wave
  const int lane = tid & 31;
  const int hi   = lane >> 4;        // lane half: A/B element j holds K = 2*hi+j
  const int n    = lane & 15;        // B/D column (output pixel) & A row (channel)
  const int cb   = wvid * 16;
  const int py   = n >> 3;
  const int cpln = (n >> 1) & 3;
  const int px   = n & 1;

  for (int g = 0; g < TBCP / 4; ++g) {
    v8f acc = {};
    const int xbase = (cb + n) * CHSTR + 4 * g + 2 * hi;
    const int wbase = (hi * BCOL + g * 16 + n) * 2;
#pragma unroll
    for (int t = 0; t < K2; ++t) {           // one 5x5 tap per K-chunk of 4
      const int ki = t / KS;
      const int kj = t - ki * KS;
      v2f a;
      a[0] = xs[xbase + ki * XW + kj];       // K = 4t + 2*hi + 0
      a[1] = xs[xbase + ki * XW + kj + 1];   // K = 4t + 2*hi + 1
      const v2f bb = *(const v2f*)(wsB + wbase + t * (2 * BCOL * 2));  // 8B load
      acc = __builtin_amdgcn_wmma_f32_16x16x4_f32(
          /*neg_a=*/false, a, /*neg_b=*/false, bb,
          /*c_mod=*/(short)0, acc, /*reuse_a=*/false, /*reuse_b=*/false);
    }
    // D layout: VGPR i, lanes 0-15 -> (m=i, n), lanes 16-31 -> (m=i+8, n-16)
    const int oh = R_ * h0 + py;
    const int ow = R_ * (w0 + 4 * g + cpln) + px;
    float* ob = out + ((size_t)b * C_ + cb + 8 * hi) * (OH_ * OW_)
                    + (size_t)oh * OW_ + ow;
#pragma unroll
    for (int i2 = 0; i2 < 8; ++i2)
      ob[(size_t)i2 * (OH_ * OW_)] = acc[i2];   // 32B-contiguous per 8 lanes
  }
}

extern "C" void kernel_launch(void* const* d_in, const int* in_sizes, int n_in,
                              void* d_out, int out_size, void* d_ws, size_t ws_size,
                              hipStream_t stream) {
  const float* x    = (const float*)d_in[0];
  const float* kern = (const float*)d_in[1];
  // d_in[2] = kernel_size (5), d_in[3] = ratio (2): hardcoded for these shapes.
  float* out = (float*)d_out;
  (void)in_sizes; (void)n_in; (void)out_size; (void)d_ws; (void)ws_size;

  dim3 grid(W_ / TBCP, H_, B_);   // (8, 64, 4)
  dim3 block(NTHR, 1, 1);
  carafe_wmma_async<<<grid, block, SMEM_BYTES, stream>>>(x, kern, out);
}